// Mamba_Model_62560493633819
// MI455X (gfx1250) — compile-verified
//
#include <hip/hip_runtime.h>
#include <math.h>

typedef __attribute__((ext_vector_type(2))) float v2f;
typedef __attribute__((ext_vector_type(8))) float v8f;

#define BB      16
#define LL      2048
#define CIN     64
#define DMODEL  128
#define DSTATE  16
#define DCONV   4
#define DINNER  256
#define DTRANK  8
#define BL      (BB * LL)   // 32768 tokens

// ---------------------------------------------------------------------------
// fp32 WMMA: D(16x16) += A(16x4) * B(4x16), full fp32 precision (CDNA5
// V_WMMA_F32_16X16X4_F32). GEMMs here total ~7.7 GFLOP -> memory/latency
// bound, so fp32 WMMA (no convert traffic, exact numerics) is the right pick.
// ---------------------------------------------------------------------------
__device__ __forceinline__ v8f wmma_f32_4(v2f a, v2f b, v8f c) {
  // 8 args: (neg_a, A, neg_b, B, c_mod, C, reuse_a, reuse_b)
  return __builtin_amdgcn_wmma_f32_16x16x4_f32(false, a, false, b,
                                               (short)0, c, false, false);
}

// Generic C(MxN) = A(MxK) * W(NxK)^T [+ bias], row-major, one wave per 16x16
// tile. A-frag: lane l -> row (l&15), K pair {0,1} (l<16) or {2,3} (l>=16),
// contiguous v2f load. B-frag mirrored from W rows. M must be /16; N guarded.
__global__ __launch_bounds__(256) void gemm_wmma_f32_kernel(
    const float* __restrict__ A, const float* __restrict__ W,
    const float* __restrict__ bias, float* __restrict__ C,
    int M, int N, int K, int ldc) {
  int wave = (int)((blockIdx.x * blockDim.x + threadIdx.x) >> 5);
  int lane = (int)(threadIdx.x & 31);
  int ntiles = (N + 15) >> 4;
  int mtiles = M >> 4;
  if (wave >= mtiles * ntiles) return;
  int tm = wave / ntiles;
  int tn = wave - tm * ntiles;
  int row0 = tm << 4, col0 = tn << 4;
  int l15   = lane & 15;
  int khalf = (lane >> 4) << 1;          // K offset of this half-wave: 0 or 2
  int ncol  = col0 + l15;
  int ncl   = (ncol < N) ? ncol : (N - 1);   // clamp; masked at store
  const float* Ap = A + (size_t)(row0 + l15) * K + khalf;
  const float* Wp = W + (size_t)ncl * K + khalf;

  v8f acc = {};
  for (int k = 0; k < K; k += 4) {
    v2f a = *(const v2f*)(Ap + k);
    v2f b = *(const v2f*)(Wp + k);
    acc = wmma_f32_4(a, b, acc);
  }

  if (ncol < N) {
    float bv = bias ? bias[ncol] : 0.0f;
    int mhi = (lane >> 4) << 3;          // lanes>=16 hold rows m+8
#pragma unroll
    for (int v = 0; v < 8; ++v)
      C[(size_t)(row0 + mhi + v) * ldc + ncol] = acc[v] + bv;
  }
}

// ---------------------------------------------------------------------------
// Depthwise causal conv1d (width 4) over time + SiLU. xm = xz[:, 0:256].
// ---------------------------------------------------------------------------
__global__ __launch_bounds__(256) void conv_silu_kernel(
    const float* __restrict__ xz, const float* __restrict__ cw,
    const float* __restrict__ cb, float* __restrict__ xm_c) {
  size_t idx = (size_t)blockIdx.x * blockDim.x + threadIdx.x;  // BL*256
  int d = (int)(idx & (DINNER - 1));
  int r = (int)(idx >> 8);
  int t = r & (LL - 1);
  float acc = cb[d];
#pragma unroll
  for (int j = 0; j < DCONV; ++j) {
    int tt = t - (DCONV - 1) + j;
    if (tt >= 0)
      acc += xz[(size_t)(r - (DCONV - 1) + j) * (2 * DINNER) + d] *
             cw[d * DCONV + j];
  }
  float sig = 1.0f / (1.0f + expf(-acc));
  xm_c[idx] = acc * sig;                 // SiLU
}

// ---------------------------------------------------------------------------
// dt = softplus(x_dbl[:, 0:8] @ w_dt^T + b_dt); K=8 -> plain VALU.
// ---------------------------------------------------------------------------
__global__ __launch_bounds__(256) void dt_proj_kernel(
    const float* __restrict__ xdbl, const float* __restrict__ w_dt,
    const float* __restrict__ b_dt, float* __restrict__ dts) {
  size_t idx = (size_t)blockIdx.x * blockDim.x + threadIdx.x;  // BL*256
  int d = (int)(idx & (DINNER - 1));
  int r = (int)(idx >> 8);
  float acc = b_dt[d];
  const float* xr = xdbl + (size_t)r * 48;
  const float* wr = w_dt + (size_t)d * DTRANK;
#pragma unroll
  for (int k = 0; k < DTRANK; ++k) acc += xr[k] * wr[k];
  dts[idx] = (acc > 20.0f) ? acc : log1pf(expf(acc));   // softplus
}

// ---------------------------------------------------------------------------
// Selective scan: 65536 independent recurrences, one lane per (b,d,s).
// 2048 waves -> full-machine occupancy for the sequential part. Per step:
// h = exp(dt*A)*h + dt*x*B;  y = sum_s h*C via 16-lane shfl_xor butterfly.
// Fused epilogue: y = (y + x*D) * silu(z).
// ---------------------------------------------------------------------------
__global__ __launch_bounds__(256) void scan_kernel(
    const float* __restrict__ dts, const float* __restrict__ xm_c,
    const float* __restrict__ xdbl, const float* __restrict__ xz,
    const float* __restrict__ A_log, const float* __restrict__ Dp,
    float* __restrict__ y) {
  int g = (int)(blockIdx.x * blockDim.x + threadIdx.x);  // 16*256*16 lanes
  int s = g & (DSTATE - 1);
  int d = (g >> 4) & (DINNER - 1);
  int b = g >> 12;
  float Ads = -expf(A_log[d * DSTATE + s]);
  float Dd  = Dp[d];
  float h   = 0.0f;
  size_t rbase = (size_t)b * LL;
  for (int t = 0; t < LL; ++t) {
    size_t r = rbase + t;
    float dtv = dts[r * DINNER + d];
    float xv  = xm_c[r * DINNER + d];
    float Bv  = xdbl[r * 48 + DTRANK + s];
    float Cv  = xdbl[r * 48 + DTRANK + DSTATE + s];
    h = expf(dtv * Ads) * h + (dtv * xv) * Bv;
    float p = h * Cv;
    p += __shfl_xor(p, 8, 16);
    p += __shfl_xor(p, 4, 16);
    p += __shfl_xor(p, 2, 16);
    p += __shfl_xor(p, 1, 16);
    if (s == 0) {
      float zv  = xz[r * (2 * DINNER) + DINNER + d];
      float sig = 1.0f / (1.0f + expf(-zv));
      y[r * DINNER + d] = (p + xv * Dd) * (zv * sig);
    }
  }
}

// ---------------------------------------------------------------------------
// out = y @ w_out^T (32768x128, K=256) fused with head dot (128->1) + bias +
// sigmoid. Block = 16 rows; 8 waves cover the 128 columns (one 16x16 WMMA
// tile each); epilogue reduces out*w_head over n via shuffles + LDS.
// ---------------------------------------------------------------------------
__global__ __launch_bounds__(256) void out_head_kernel(
    const float* __restrict__ y, const float* __restrict__ Wout,
    const float* __restrict__ wh, const float* __restrict__ bh,
    float* __restrict__ out) {
  __shared__ float partial[8][16];
  int wv   = (int)(threadIdx.x >> 5);
  int lane = (int)(threadIdx.x & 31);
  int row0 = (int)blockIdx.x << 4;
  int l15   = lane & 15;
  int khalf = (lane >> 4) << 1;
  int n     = (wv << 4) + l15;                 // column 0..127
  const float* Ap = y + (size_t)(row0 + l15) * DINNER + khalf;
  const float* Wp = Wout + (size_t)n * DINNER + khalf;

  v8f acc = {};
  for (int k = 0; k < DINNER; k += 4) {
    v2f a = *(const v2f*)(Ap + k);
    v2f b = *(const v2f*)(Wp + k);
    acc = wmma_f32_4(a, b, acc);
  }

  float whn = wh[n];
  float pm[8];
#pragma unroll
  for (int v = 0; v < 8; ++v) {
    pm[v] = acc[v] * whn;
    pm[v] += __shfl_xor(pm[v], 1, 16);
    pm[v] += __shfl_xor(pm[v], 2, 16);
    pm[v] += __shfl_xor(pm[v], 4, 16);
    pm[v] += __shfl_xor(pm[v], 8, 16);
  }
  if (l15 == 0) {
    int mhi = (lane >> 4) << 3;
#pragma unroll
    for (int v = 0; v < 8; ++v) partial[wv][mhi + v] = pm[v];
  }
  __syncthreads();
  if (threadIdx.x < 16) {
    float sum = bh[0];
#pragma unroll
    for (int w = 0; w < 8; ++w) sum += partial[w][threadIdx.x];
    out[row0 + threadIdx.x] = 1.0f / (1.0f + expf(-sum));
  }
}

// ---------------------------------------------------------------------------
extern "C" void kernel_launch(void* const* d_in, const int* in_sizes, int n_in,
                              void* d_out, int out_size, void* d_ws,
                              size_t ws_size, hipStream_t stream) {
  (void)in_sizes; (void)n_in; (void)out_size; (void)ws_size;
  const float* x      = (const float*)d_in[0];
  const float* w_pin  = (const float*)d_in[1];
  const float* b_pin  = (const float*)d_in[2];
  const float* w_inp  = (const float*)d_in[3];
  const float* conv_w = (const float*)d_in[4];
  const float* conv_b = (const float*)d_in[5];
  const float* w_xp   = (const float*)d_in[6];
  const float* w_dt   = (const float*)d_in[7];
  const float* b_dt   = (const float*)d_in[8];
  const float* A_log  = (const float*)d_in[9];
  const float* D_par  = (const float*)d_in[10];
  const float* w_outp = (const float*)d_in[11];
  const float* w_head = (const float*)d_in[12];
  const float* b_head = (const float*)d_in[13];
  float* out = (float*)d_out;

  // Workspace layout (floats). h is dead after in_proj -> aliased under y.
  float* ws   = (float*)d_ws;
  float* y    = ws;                            // BL*256  (scan output)
  float* h    = ws;                            // BL*128  (aliased, dead early)
  float* xz   = ws + (size_t)BL * 256;         // BL*512
  float* xm_c = xz + (size_t)BL * 512;         // BL*256
  float* xdbl = xm_c + (size_t)BL * 256;       // BL*48 (40 used, ld=48)
  float* dts  = xdbl + (size_t)BL * 48;        // BL*256

  // 1) h = x @ w_proj_in^T + b        (32768x128, K=64)
  {
    int M = BL, N = DMODEL, K = CIN;
    int waves = (M >> 4) * ((N + 15) >> 4);
    gemm_wmma_f32_kernel<<<(waves * 32 + 255) / 256, 256, 0, stream>>>(
        x, w_pin, b_pin, h, M, N, K, N);
  }
  // 2) xz = h @ w_in_proj^T           (32768x512, K=128)
  {
    int M = BL, N = 2 * DINNER, K = DMODEL;
    int waves = (M >> 4) * ((N + 15) >> 4);
    gemm_wmma_f32_kernel<<<(waves * 32 + 255) / 256, 256, 0, stream>>>(
        h, w_inp, nullptr, xz, M, N, K, N);
  }
  // 3) depthwise causal conv + SiLU -> xm_c
  conv_silu_kernel<<<(size_t)BL * DINNER / 256, 256, 0, stream>>>(
      xz, conv_w, conv_b, xm_c);
  // 4) x_dbl = xm_c @ w_x_proj^T      (32768x40, K=256, ldc=48)
  {
    int M = BL, N = DTRANK + 2 * DSTATE, K = DINNER;
    int waves = (M >> 4) * ((N + 15) >> 4);
    gemm_wmma_f32_kernel<<<(waves * 32 + 255) / 256, 256, 0, stream>>>(
        xm_c, w_xp, nullptr, xdbl, M, N, K, 48);
  }
  // 5) dt = softplus(x_dbl[:, :8] @ w_dt^T + b_dt)
  dt_proj_kernel<<<(size_t)BL * DINNER / 256, 256, 0, stream>>>(
      xdbl, w_dt, b_dt, dts);
  // 6) selective scan, fused (+x*D)*silu(z)
  scan_kernel<<<(BB * DINNER * DSTATE) / 256, 256, 0, stream>>>(
      dts, xm_c, xdbl, xz, A_log, D_par, y);
  // 7) out_proj + head + sigmoid -> d_out (B,L)
  out_head_kernel<<<BL / 16, 256, 0, stream>>>(y, w_outp, w_head, b_head, out);
}